// STGCN_89300960018576
// MI455X (gfx1250) — compile-verified
//
#include <hip/hip_runtime.h>
#include <hip/hip_bf16.h>
#include <math.h>

// ---------------------------------------------------------------------------
// STGCN for MI455X (gfx1250, wave32).
// Hot path: lstm_in(192x32000) @ Wih^T(32000x512) via V_WMMA_F32_16X16X4_F32.
// Register tiling: 2 M-tiles x 4 N-tiles per wave (6 b64 loads : 8 WMMAs),
// K split 32 ways for occupancy (1536 waves). LSTM recurrence uses WMMA on
// LDS-resident h-state in a single persistent block.
// ---------------------------------------------------------------------------

typedef float v2f __attribute__((ext_vector_type(2)));
typedef float v8f __attribute__((ext_vector_type(8)));

#define B_    16
#define T_    12
#define N_    2000
#define E_    32000
#define F_    16
#define HID_  128
#define G4_   512            // 4*HID
#define K_    32000          // N*F (LSTM input width)
#define BT_   192            // B*T
#define NNZ_  (E_ + N_)      // edges + self loops
#define KSPLIT_ 32
#define KCHUNK_ (K_ / KSPLIT_)   // 1000

__device__ __forceinline__ float sigmoidf_(float x) { return 1.0f / (1.0f + expf(-x)); }

// ---------------- graph preprocessing: CSR by destination ------------------

__global__ void init_deg_kernel(int* __restrict__ deg, int* __restrict__ fill) {
    int n = blockIdx.x * blockDim.x + threadIdx.x;
    if (n < N_) { deg[n] = 1; fill[n] = 0; }   // deg starts at 1: self loop
}

__global__ void hist_kernel(const long long* __restrict__ ei, int* __restrict__ deg) {
    int e = blockIdx.x * blockDim.x + threadIdx.x;
    if (e < E_) atomicAdd(&deg[(int)ei[E_ + e]], 1);
}

__global__ void dinv_kernel(const int* __restrict__ deg, float* __restrict__ dinv) {
    int n = blockIdx.x * blockDim.x + threadIdx.x;
    if (n < N_) dinv[n] = rsqrtf((float)deg[n]);   // deg >= 1 always
}

__global__ void scan_kernel(const int* __restrict__ deg, int* __restrict__ rowptr) {
    if (blockIdx.x == 0 && threadIdx.x == 0) {
        int s = 0;
        for (int n = 0; n < N_; ++n) { rowptr[n] = s; s += deg[n]; }
        rowptr[N_] = s;
    }
}

__global__ void fill_kernel(const long long* __restrict__ ei,
                            const int* __restrict__ rowptr, int* __restrict__ fill,
                            const float* __restrict__ dinv,
                            int* __restrict__ col, float* __restrict__ wgt) {
    int e = blockIdx.x * blockDim.x + threadIdx.x;
    if (e >= NNZ_) return;
    int s, d;
    if (e < E_) { s = (int)ei[e]; d = (int)ei[E_ + e]; }
    else        { s = d = e - E_; }                      // self loop
    int pos = rowptr[d] + atomicAdd(&fill[d], 1);
    col[pos] = s;
    wgt[pos] = dinv[s] * dinv[d];
}

// ---------------- GCN layer 1 (scalar aggregate, rank-1 expand) ------------

__global__ void gcn1_kernel(const float* __restrict__ x,
                            const int* __restrict__ rowptr, const int* __restrict__ col,
                            const float* __restrict__ wgt,
                            const float* __restrict__ W1, const float* __restrict__ b1,
                            float* __restrict__ h1) {
    int i = blockIdx.x * blockDim.x + threadIdx.x;
    if (i >= BT_ * N_) return;
    int n = i % N_, bt = i / N_;
    const float* xrow = x + (size_t)bt * N_;
    float s = 0.0f;
    int e1 = rowptr[n + 1];
    for (int e = rowptr[n]; e < e1; ++e) s += wgt[e] * xrow[col[e]];
    float4* out = (float4*)(h1 + (size_t)i * F_);
#pragma unroll
    for (int q = 0; q < 4; ++q) {
        float4 w  = ((const float4*)W1)[q];
        float4 bb = ((const float4*)b1)[q];
        float4 v;
        v.x = fmaxf(s * w.x + bb.x, 0.0f);
        v.y = fmaxf(s * w.y + bb.y, 0.0f);
        v.z = fmaxf(s * w.z + bb.z, 0.0f);
        v.w = fmaxf(s * w.w + bb.w, 0.0f);
        out[q] = v;
    }
}

// ------ GCN layer 2 (16-wide aggregate, fused 16x16 matvec, write lstm_in) -

__global__ void gcn2_kernel(const float* __restrict__ h1,
                            const int* __restrict__ rowptr, const int* __restrict__ col,
                            const float* __restrict__ wgt,
                            const float* __restrict__ W2, const float* __restrict__ b2,
                            float* __restrict__ lstm_in) {
    __shared__ float sW2[256];
    __shared__ float sb2[16];
    for (int q = threadIdx.x; q < 256; q += blockDim.x) sW2[q] = W2[q];
    if (threadIdx.x < 16) sb2[threadIdx.x] = b2[threadIdx.x];
    __syncthreads();

    int i = blockIdx.x * blockDim.x + threadIdx.x;
    if (i >= BT_ * N_) return;
    int n = i % N_, bt = i / N_;
    float acc[16];
#pragma unroll
    for (int f = 0; f < 16; ++f) acc[f] = 0.0f;

    int e1 = rowptr[n + 1];
    for (int e = rowptr[n]; e < e1; ++e) {
        float w = wgt[e];
        const float4* hr = (const float4*)(h1 + ((size_t)bt * N_ + col[e]) * F_);
#pragma unroll
        for (int q = 0; q < 4; ++q) {
            float4 hv = hr[q];
            acc[4 * q + 0] = fmaf(w, hv.x, acc[4 * q + 0]);
            acc[4 * q + 1] = fmaf(w, hv.y, acc[4 * q + 1]);
            acc[4 * q + 2] = fmaf(w, hv.z, acc[4 * q + 2]);
            acc[4 * q + 3] = fmaf(w, hv.w, acc[4 * q + 3]);
        }
    }
    float* out = lstm_in + (size_t)bt * K_ + (size_t)n * F_;
#pragma unroll
    for (int fo = 0; fo < 16; ++fo) {
        float v = sb2[fo];
#pragma unroll
        for (int fi = 0; fi < 16; ++fi) v = fmaf(acc[fi], sW2[fi * 16 + fo], v);
        out[fo] = fmaxf(v, 0.0f);
    }
}

// ---------- big GEMM: Gpart[k][bt][g] = A(192x32000) x Wih^T tile ----------
// FP32 WMMA 16x16x4. A frag (16x4 f32): lane = M (lane&15), K-half = lane>>4,
// 2 consecutive K-values per lane -> one b64 load. B[k][n] = Wih[n][k], same
// per-lane addressing against Wih rows. D layout: vgpr r -> M = r + 8*(lane>>4).
// Each wave: 2 M-tiles x 4 N-tiles -> 6 loads feed 8 WMMAs per k-step.

__global__ void __launch_bounds__(256) gemm_kernel(const float* __restrict__ A,
                                                   const float* __restrict__ Wih,
                                                   float* __restrict__ Gpart) {
    int wid  = blockIdx.x * (blockDim.x >> 5) + (threadIdx.x >> 5);
    int lane = threadIdx.x & 31;
    int kidx = wid % KSPLIT_;
    int ng   = (wid / KSPLIT_) % 8;       // 8 groups of 64 gate columns
    int mg   = wid / (KSPLIT_ * 8);       // 6 groups of 32 rows
    int m0 = mg * 32;
    int n0 = ng * 64;
    int r  = lane & 15;
    int kh = (lane >> 4) * 2;

    const float* ap0 = A   + (size_t)(m0 +  0 + r) * K_ + kh;
    const float* ap1 = A   + (size_t)(m0 + 16 + r) * K_ + kh;
    const float* bp0 = Wih + (size_t)(n0 +  0 + r) * K_ + kh;
    const float* bp1 = Wih + (size_t)(n0 + 16 + r) * K_ + kh;
    const float* bp2 = Wih + (size_t)(n0 + 32 + r) * K_ + kh;
    const float* bp3 = Wih + (size_t)(n0 + 48 + r) * K_ + kh;

    v8f acc00 = {}, acc01 = {}, acc02 = {}, acc03 = {};
    v8f acc10 = {}, acc11 = {}, acc12 = {}, acc13 = {};
    int k0 = kidx * KCHUNK_;
    for (int k = k0; k < k0 + KCHUNK_; k += 4) {
        v2f a0 = *(const v2f*)(ap0 + k);
        v2f a1 = *(const v2f*)(ap1 + k);
        v2f b0 = *(const v2f*)(bp0 + k);
        v2f b1 = *(const v2f*)(bp1 + k);
        v2f b2 = *(const v2f*)(bp2 + k);
        v2f b3 = *(const v2f*)(bp3 + k);
        acc00 = __builtin_amdgcn_wmma_f32_16x16x4_f32(false, a0, false, b0, (short)0, acc00, false, false);
        acc01 = __builtin_amdgcn_wmma_f32_16x16x4_f32(false, a0, false, b1, (short)0, acc01, false, false);
        acc02 = __builtin_amdgcn_wmma_f32_16x16x4_f32(false, a0, false, b2, (short)0, acc02, false, false);
        acc03 = __builtin_amdgcn_wmma_f32_16x16x4_f32(false, a0, false, b3, (short)0, acc03, false, false);
        acc10 = __builtin_amdgcn_wmma_f32_16x16x4_f32(false, a1, false, b0, (short)0, acc10, false, false);
        acc11 = __builtin_amdgcn_wmma_f32_16x16x4_f32(false, a1, false, b1, (short)0, acc11, false, false);
        acc12 = __builtin_amdgcn_wmma_f32_16x16x4_f32(false, a1, false, b2, (short)0, acc12, false, false);
        acc13 = __builtin_amdgcn_wmma_f32_16x16x4_f32(false, a1, false, b3, (short)0, acc13, false, false);
    }
    int rowh = (lane >> 4) * 8;
#pragma unroll
    for (int rr = 0; rr < 8; ++rr) {
        size_t base0 = ((size_t)kidx * BT_ + (size_t)(m0 +      rowh + rr)) * G4_ + n0 + r;
        size_t base1 = ((size_t)kidx * BT_ + (size_t)(m0 + 16 + rowh + rr)) * G4_ + n0 + r;
        Gpart[base0 +  0] = acc00[rr];
        Gpart[base0 + 16] = acc01[rr];
        Gpart[base0 + 32] = acc02[rr];
        Gpart[base0 + 48] = acc03[rr];
        Gpart[base1 +  0] = acc10[rr];
        Gpart[base1 + 16] = acc11[rr];
        Gpart[base1 + 32] = acc12[rr];
        Gpart[base1 + 48] = acc13[rr];
    }
}

__global__ void reduce_kernel(const float* __restrict__ Gpart, float* __restrict__ G) {
    int i = blockIdx.x * blockDim.x + threadIdx.x;
    if (i >= BT_ * G4_) return;
    float s = 0.0f;
#pragma unroll
    for (int k = 0; k < KSPLIT_; ++k) s += Gpart[(size_t)k * (BT_ * G4_) + i];
    G[i] = s;
}

// -------- LSTM recurrence: single persistent block, WMMA on LDS h ----------
// 8 waves; wave w owns gate columns [w*64, w*64+64). Per step:
// gates(16x512) = h(16x128) @ Whh^T via WMMA f32 16x16x4, then fused gate math.

__global__ void __launch_bounds__(256) lstm_kernel(const float* __restrict__ G,
                                                   const float* __restrict__ Whh,
                                                   const float* __restrict__ bih,
                                                   const float* __restrict__ bhh,
                                                   float* __restrict__ hlast) {
    __shared__ float sh_h[B_ * HID_];
    __shared__ float sh_c[B_ * HID_];
    __shared__ float sh_g[B_ * G4_];
    int tid = threadIdx.x;
    for (int i = tid; i < B_ * HID_; i += 256) { sh_h[i] = 0.0f; sh_c[i] = 0.0f; }
    __syncthreads();

    int wave = tid >> 5, lane = tid & 31;
    int r = lane & 15;
    int kh = (lane >> 4) * 2;
    int rowh = (lane >> 4) * 8;

    for (int t = 0; t < T_; ++t) {
#pragma unroll
        for (int j = 0; j < 4; ++j) {
            int n0 = wave * 64 + j * 16;
            const float* bp = Whh + (size_t)(n0 + r) * HID_ + kh;
            v8f acc = {};
#pragma unroll
            for (int k = 0; k < HID_; k += 4) {
                v2f a = *(const v2f*)&sh_h[r * HID_ + k + kh];
                v2f b = *(const v2f*)(bp + k);
                acc = __builtin_amdgcn_wmma_f32_16x16x4_f32(false, a, false, b, (short)0, acc, false, false);
            }
#pragma unroll
            for (int rr = 0; rr < 8; ++rr)
                sh_g[(rowh + rr) * G4_ + n0 + r] = acc[rr];
        }
        __syncthreads();

        for (int p = tid; p < B_ * HID_; p += 256) {
            int b = p / HID_, h = p % HID_;
            const float* gRow = G + (size_t)(b * T_ + t) * G4_;
            int i0 = b * G4_ + h;
            float gi = sh_g[i0           ] + gRow[h           ] + bih[h           ] + bhh[h           ];
            float gf = sh_g[i0 +     HID_] + gRow[h +     HID_] + bih[h +     HID_] + bhh[h +     HID_];
            float gg = sh_g[i0 + 2 * HID_] + gRow[h + 2 * HID_] + bih[h + 2 * HID_] + bhh[h + 2 * HID_];
            float go = sh_g[i0 + 3 * HID_] + gRow[h + 3 * HID_] + bih[h + 3 * HID_] + bhh[h + 3 * HID_];
            float c = sh_c[p];
            c = sigmoidf_(gf) * c + sigmoidf_(gi) * tanhf(gg);
            sh_h[p] = sigmoidf_(go) * tanhf(c);
            sh_c[p] = c;
        }
        __syncthreads();
    }
    for (int i = tid; i < B_ * HID_; i += 256) hlast[i] = sh_h[i];
}

// ------------------ final FC + residual ------------------------------------

__global__ void fc_kernel(const float* __restrict__ hlast, const float* __restrict__ Wfc,
                          const float* __restrict__ bfc, const float* __restrict__ x,
                          float* __restrict__ out) {
    int i = blockIdx.x * blockDim.x + threadIdx.x;
    if (i >= B_ * N_) return;
    int n = i % N_, b = i / N_;
    const float4* hv = (const float4*)(hlast + (size_t)b * HID_);
    const float4* wv = (const float4*)(Wfc + (size_t)n * HID_);
    float s = 0.0f;
#pragma unroll 8
    for (int q = 0; q < HID_ / 4; ++q) {
        float4 a = hv[q], w = wv[q];
        s += a.x * w.x + a.y * w.y + a.z * w.z + a.w * w.w;
    }
    out[i] = s + bfc[n] + x[((size_t)b * T_ + (T_ - 1)) * N_ + n];
}

// ---------------------------------------------------------------------------

extern "C" void kernel_launch(void* const* d_in, const int* in_sizes, int n_in,
                              void* d_out, int out_size, void* d_ws, size_t ws_size,
                              hipStream_t stream) {
    const float*     x    = (const float*)d_in[0];
    const long long* ei   = (const long long*)d_in[1];   // int64 edge_index (2, E)
    const float*     W1   = (const float*)d_in[2];
    const float*     b1   = (const float*)d_in[3];
    const float*     W2   = (const float*)d_in[4];
    const float*     b2   = (const float*)d_in[5];
    const float*     Wih  = (const float*)d_in[6];
    const float*     Whh  = (const float*)d_in[7];
    const float*     bih  = (const float*)d_in[8];
    const float*     bhh  = (const float*)d_in[9];
    const float*     Wfc  = (const float*)d_in[10];
    const float*     bfc  = (const float*)d_in[11];
    float*           out  = (float*)d_out;

    char* ws = (char*)d_ws;
    size_t off = 0;
    auto carve = [&](size_t bytes) -> void* {
        void* p = ws + off;
        off = (off + bytes + 255) & ~(size_t)255;
        return p;
    };
    int*   deg     = (int*)  carve(sizeof(int) * N_);
    int*   fill    = (int*)  carve(sizeof(int) * N_);
    int*   rowptr  = (int*)  carve(sizeof(int) * (N_ + 1));
    float* dinv    = (float*)carve(sizeof(float) * N_);
    int*   col     = (int*)  carve(sizeof(int) * NNZ_);
    float* wgt     = (float*)carve(sizeof(float) * NNZ_);
    float* h1      = (float*)carve(sizeof(float) * (size_t)BT_ * N_ * F_);   // 24.6 MB
    float* lstm_in = (float*)carve(sizeof(float) * (size_t)BT_ * K_);        // 24.6 MB
    float* Gpart   = (float*)carve(sizeof(float) * (size_t)KSPLIT_ * BT_ * G4_); // 12.6 MB
    float* G       = (float*)carve(sizeof(float) * (size_t)BT_ * G4_);
    float* hlast   = (float*)carve(sizeof(float) * B_ * HID_);

    // --- graph preprocessing ---
    init_deg_kernel<<<(N_ + 255) / 256, 256, 0, stream>>>(deg, fill);
    hist_kernel<<<(E_ + 255) / 256, 256, 0, stream>>>(ei, deg);
    dinv_kernel<<<(N_ + 255) / 256, 256, 0, stream>>>(deg, dinv);
    scan_kernel<<<1, 32, 0, stream>>>(deg, rowptr);
    fill_kernel<<<(NNZ_ + 255) / 256, 256, 0, stream>>>(ei, rowptr, fill, dinv, col, wgt);

    // --- GCN layers ---
    gcn1_kernel<<<(BT_ * N_ + 255) / 256, 256, 0, stream>>>(x, rowptr, col, wgt, W1, b1, h1);
    gcn2_kernel<<<(BT_ * N_ + 255) / 256, 256, 0, stream>>>(h1, rowptr, col, wgt, W2, b2, lstm_in);

    // --- batched LSTM input GEMM (WMMA fp32), 2x4 register tile, K split 32 ---
    gemm_kernel<<<(6 * 8 * KSPLIT_) / 8, 256, 0, stream>>>(lstm_in, Wih, Gpart);
    reduce_kernel<<<(BT_ * G4_ + 255) / 256, 256, 0, stream>>>(Gpart, G);

    // --- LSTM recurrence (persistent single block, WMMA on LDS state) ---
    lstm_kernel<<<1, 256, 0, stream>>>(G, Whh, bih, bhh, hlast);

    // --- final FC + residual ---
    fc_kernel<<<(B_ * N_ + 255) / 256, 256, 0, stream>>>(hlast, Wfc, bfc, x, out);
}